// RandomWalkEmbed_11312943857862
// MI455X (gfx1250) — compile-verified
//
#include <hip/hip_runtime.h>
#include <math.h>

typedef __attribute__((ext_vector_type(2))) float v2f;
typedef __attribute__((ext_vector_type(8))) float v8f;

constexpr int NP = 16;
constexpr int NG = 64;
constexpr int C  = 2048;
constexpr int HW = 32;          // 8*4 spatial
constexpr int NPAIR = NP * NG;  // 1024

// ---------------------------------------------------------------------------
// K0a: inverse L2 norm over channel dim for each (sample, spatial) column.
// inv[col] = 1 / max(||x[samp, :, r]||_2, 1e-12)
// ---------------------------------------------------------------------------
__global__ __launch_bounds__(256) void invnorm_kernel(const float* __restrict__ x,
                                                      float* __restrict__ inv) {
    const int col  = blockIdx.x;      // samp*HW + r
    const int samp = col >> 5;
    const int r    = col & 31;
    const float* base = x + (size_t)samp * C * HW + r;
    float sum = 0.f;
    for (int c = threadIdx.x; c < C; c += 256) {
        float v = base[(size_t)c * HW];
        sum += v * v;
    }
    __shared__ float red[256];
    red[threadIdx.x] = sum;
    __syncthreads();
    for (int s = 128; s > 0; s >>= 1) {
        if ((int)threadIdx.x < s) red[threadIdx.x] += red[threadIdx.x + s];
        __syncthreads();
    }
    if (threadIdx.x == 0) {
        float n = sqrtf(red[0]);
        inv[col] = 1.0f / fmaxf(n, 1e-12f);
    }
}

// ---------------------------------------------------------------------------
// K0b: gbar[g, c] = mean_q gallery[g, c, q]
// ---------------------------------------------------------------------------
__global__ __launch_bounds__(256) void gbar_kernel(const float* __restrict__ gallery,
                                                   float* __restrict__ gbar) {
    const int idx = blockIdx.x * 256 + threadIdx.x;   // g*C + c
    if (idx >= NG * C) return;
    const float* base = gallery + (size_t)idx * HW;
    float s = 0.f;
#pragma unroll
    for (int q = 0; q < HW; ++q) s += base[q];
    gbar[idx] = s * (1.0f / HW);
}

// ---------------------------------------------------------------------------
// K1: corr[p,g,q,r] = (sum_c g[g,c,q] * p[p,c,r]) * ginv[g,q] * pinv[p,r]
// 1 block per (p,g) pair; 4 waves = 2x2 tiles of 16x16; K=2048 via wmma K=4.
// ---------------------------------------------------------------------------
__global__ __launch_bounds__(128) void corr_wmma_kernel(const float* __restrict__ probe,
                                                        const float* __restrict__ gallery,
                                                        const float* __restrict__ pinv,
                                                        const float* __restrict__ ginv,
                                                        float* __restrict__ corr) {
    const int pair = blockIdx.x;            // 0..1023
    const int pIdx = pair >> 6;
    const int gIdx = pair & 63;
    const int wave = threadIdx.x >> 5;      // 0..3
    const int lane = threadIdx.x & 31;
    const int q0   = (wave >> 1) * 16;      // tile M base (gallery q)
    const int r0   = (wave & 1) * 16;       // tile N base (probe r)
    const int half = lane >> 4;             // k-half select
    const int l16  = lane & 15;

    // A[m,k] = gallery[gIdx, k, q0+m]   (stride HW between k)
    // B[k,n] = probe  [pIdx, k, r0+n]
    const float* gBase = gallery + (size_t)gIdx * C * HW + (q0 + l16) + (size_t)(2 * half) * HW;
    const float* pBase = probe   + (size_t)pIdx * C * HW + (r0 + l16) + (size_t)(2 * half) * HW;

    v8f acc = {};
#pragma unroll 4
    for (int k = 0; k < C; k += 4) {
        v2f a, b;
        a.x = gBase[(size_t)k * HW];
        a.y = gBase[(size_t)k * HW + HW];
        b.x = pBase[(size_t)k * HW];
        b.y = pBase[(size_t)k * HW + HW];
        acc = __builtin_amdgcn_wmma_f32_16x16x4_f32(false, a, false, b,
                                                    (short)0, acc, false, false);
    }

    const float sc_n = pinv[pIdx * HW + r0 + l16];   // per-lane column scale
    float* out = corr + (size_t)pair * HW * HW;
#pragma unroll
    for (int j = 0; j < 8; ++j) {
        const int m = j + 8 * half;
        const float v = acc[j] * ginv[gIdx * HW + q0 + m] * sc_n;
        out[(q0 + m) * HW + (r0 + l16)] = v;
    }
}

// ---------------------------------------------------------------------------
// K2: softmax over r of corr[pair, q, :], then s[pair, r] = sum_q att[q, r].
// One wave per pair; lane = q.
// ---------------------------------------------------------------------------
__global__ __launch_bounds__(32) void softmax_s_kernel(const float* __restrict__ corr,
                                                       float* __restrict__ sOut) {
    const int pair = blockIdx.x;
    const int q    = threadIdx.x;          // 0..31
    const float* row = corr + ((size_t)pair * HW + q) * HW;

    float v[HW];
    float mx = -INFINITY;
#pragma unroll
    for (int r = 0; r < HW; ++r) { v[r] = row[r]; mx = fmaxf(mx, v[r]); }
    float sum = 0.f;
#pragma unroll
    for (int r = 0; r < HW; ++r) { v[r] = __expf(v[r] - mx); sum += v[r]; }
    const float invs = 1.0f / sum;

    __shared__ float att[HW][HW + 1];
#pragma unroll
    for (int r = 0; r < HW; ++r) att[q][r] = v[r] * invs;
    __syncthreads();

    const int r = q;
    float colsum = 0.f;
#pragma unroll
    for (int qq = 0; qq < HW; ++qq) colsum += att[qq][r];
    sOut[(size_t)pair * HW + r] = colsum;
}

// ---------------------------------------------------------------------------
// K3: d[(p*NG+g), c] = ( (1/HW) * sum_r probe[p,c,r]*s[p,g,r] - gbar[g,c] )^2
// Per-probe GEMM M=C, N=NG, K=HW using wmma f32 16x16x4. 1 tile per wave.
// ---------------------------------------------------------------------------
__global__ __launch_bounds__(128) void pooled_wmma_kernel(const float* __restrict__ probe,
                                                          const float* __restrict__ sMat,
                                                          const float* __restrict__ gbar,
                                                          float* __restrict__ dOut) {
    const int wave = threadIdx.x >> 5;
    const int lane = threadIdx.x & 31;
    const int tile = blockIdx.x * 4 + wave;     // 0 .. NP*512-1
    const int pIdx = tile >> 9;                 // 512 tiles per probe
    const int tIn  = tile & 511;
    const int c0   = (tIn >> 2) * 16;           // 128 c-tiles
    const int g0   = (tIn & 3) * 16;            // 4 g-tiles
    const int half = lane >> 4;
    const int l16  = lane & 15;

    // A[m,k] = probe[pIdx, c0+m, k]  (k contiguous)
    // B[k,n] = sMat [pIdx*NG + g0+n, k] (k contiguous)
    const float* aBase = probe + ((size_t)pIdx * C + (c0 + l16)) * HW + 2 * half;
    const float* bBase = sMat  + ((size_t)(pIdx * NG) + (g0 + l16)) * HW + 2 * half;

    v8f acc = {};
#pragma unroll
    for (int k = 0; k < HW; k += 4) {
        v2f a = *(const v2f*)(aBase + k);
        v2f b = *(const v2f*)(bBase + k);
        acc = __builtin_amdgcn_wmma_f32_16x16x4_f32(false, a, false, b,
                                                    (short)0, acc, false, false);
    }

    const int g = g0 + l16;                     // N index
#pragma unroll
    for (int j = 0; j < 8; ++j) {
        const int c = c0 + j + 8 * half;        // M index
        const float v = acc[j] * (1.0f / HW) - gbar[(size_t)g * C + c];
        dOut[((size_t)(pIdx * NG) + g) * C + c] = v * v;
    }
}

// ---------------------------------------------------------------------------
// K4: BN batch stats over the 1024-pair batch.
// mu[c], aScale[c] = gamma[c] * rsqrt(var[c] + 1e-5)
// ---------------------------------------------------------------------------
__global__ __launch_bounds__(256) void bn_stats_kernel(const float* __restrict__ d,
                                                       const float* __restrict__ gamma,
                                                       float* __restrict__ mu,
                                                       float* __restrict__ aScale) {
    const int c = blockIdx.x * 256 + threadIdx.x;
    if (c >= C) return;
    float sum = 0.f, sumsq = 0.f;
    for (int i = 0; i < NPAIR; ++i) {
        float v = d[(size_t)i * C + c];
        sum += v;
        sumsq += v * v;
    }
    const float m   = sum * (1.0f / NPAIR);
    const float var = fmaxf(sumsq * (1.0f / NPAIR) - m * m, 0.0f);
    mu[c]     = m;
    aScale[c] = gamma[c] * rsqrtf(var + 1e-5f);
}

// ---------------------------------------------------------------------------
// K5: logits[pair, k] = sum_c ((d-mu)*aScale + beta) * w[k,c] + b[k]
// ---------------------------------------------------------------------------
__global__ __launch_bounds__(256) void logits_kernel(const float* __restrict__ d,
                                                     const float* __restrict__ mu,
                                                     const float* __restrict__ aScale,
                                                     const float* __restrict__ beta,
                                                     const float* __restrict__ w,
                                                     const float* __restrict__ b,
                                                     float* __restrict__ out) {
    const int pair = blockIdx.x;
    float a0 = 0.f, a1 = 0.f;
    for (int c = threadIdx.x; c < C; c += 256) {
        const float bn = (d[(size_t)pair * C + c] - mu[c]) * aScale[c] + beta[c];
        a0 += bn * w[c];
        a1 += bn * w[C + c];
    }
    __shared__ float r0[256], r1[256];
    r0[threadIdx.x] = a0;
    r1[threadIdx.x] = a1;
    __syncthreads();
    for (int s = 128; s > 0; s >>= 1) {
        if ((int)threadIdx.x < s) {
            r0[threadIdx.x] += r0[threadIdx.x + s];
            r1[threadIdx.x] += r1[threadIdx.x + s];
        }
        __syncthreads();
    }
    if (threadIdx.x == 0) {
        out[pair * 2 + 0] = r0[0] + b[0];
        out[pair * 2 + 1] = r1[0] + b[1];
    }
}

// ---------------------------------------------------------------------------
extern "C" void kernel_launch(void* const* d_in, const int* in_sizes, int n_in,
                              void* d_out, int out_size, void* d_ws, size_t ws_size,
                              hipStream_t stream) {
    const float* probe   = (const float*)d_in[0];  // [16,2048,8,4]
    const float* gallery = (const float*)d_in[1];  // [64,2048,8,4]
    const float* gamma   = (const float*)d_in[2];  // [2048]
    const float* beta    = (const float*)d_in[3];  // [2048]
    const float* cls_w   = (const float*)d_in[4];  // [2,2048]
    const float* cls_b   = (const float*)d_in[5];  // [2]
    float* out = (float*)d_out;                    // [16,64,2]

    float* ws   = (float*)d_ws;
    float* pinv = ws;                      // 512
    float* ginv = pinv + NP * HW;          // 2048
    float* gbar = ginv + NG * HW;          // 131072
    float* sMat = gbar + NG * C;           // 32768
    float* mu   = sMat + NPAIR * HW;       // 2048
    float* aSc  = mu + C;                  // 2048
    float* corr = aSc + C;                 // 1,048,576
    float* dbuf = corr + (size_t)NPAIR * HW * HW;  // 2,097,152
    // total ~12.7 MB of workspace

    invnorm_kernel<<<NP * HW, 256, 0, stream>>>(probe, pinv);
    invnorm_kernel<<<NG * HW, 256, 0, stream>>>(gallery, ginv);
    gbar_kernel<<<(NG * C) / 256, 256, 0, stream>>>(gallery, gbar);
    corr_wmma_kernel<<<NPAIR, 128, 0, stream>>>(probe, gallery, pinv, ginv, corr);
    softmax_s_kernel<<<NPAIR, 32, 0, stream>>>(corr, sMat);
    pooled_wmma_kernel<<<NP * 128, 128, 0, stream>>>(probe, sMat, gbar, dbuf);
    bn_stats_kernel<<<C / 256, 256, 0, stream>>>(dbuf, gamma, mu, aSc);
    logits_kernel<<<NPAIR, 256, 0, stream>>>(dbuf, mu, aSc, beta, cls_w, cls_b, out);
}